// VMFV1_10376640987792
// MI455X (gfx1250) — compile-verified
//
#include <hip/hip_runtime.h>
#include <math.h>
#include <stdint.h>

// ---------------- configuration ----------------
#define ROW_D      512            // feature dim (fixed by reference)
#define TAB_K      65536          // inverse-CDF bucket count (power of 2)
#define SCAN_BLK   1024
#define MAX_PARTS  1024           // >= ceil(n/SCAN_BLK) (196 for n=199,999)

// ---------------- CDNA5 async-to-LDS availability ----------------
#if defined(__HIP_DEVICE_COMPILE__) && __has_builtin(__builtin_amdgcn_global_load_async_to_lds_b128)
#define VMF_ASYNC_LDS 1
#else
#define VMF_ASYNC_LDS 0
#endif

#if VMF_ASYNC_LDS
// builtin signature (probe-confirmed by hipcc diagnostic): param0 = v4i addrspace(1)*,
// param1 = v4i addrspace(3)*, then two integer immediates (offset, cpol).
typedef int v4i __attribute__((__vector_size__(16)));
typedef __attribute__((address_space(1))) v4i* p1_v4i;
typedef __attribute__((address_space(3))) v4i* p3_v4i;
// generic -> as1: identical 64-bit value. generic LDS -> as3: low 32 bits are the LDS offset.
#define VMF_AS1(p) ((p1_v4i)(uintptr_t)(p))
#define VMF_AS3(p) ((p3_v4i)(uint32_t)(uintptr_t)(p))
#endif

// ---------------- device helpers ----------------
__device__ __forceinline__ float vmf_wave_sum(float v) {
    v += __shfl_xor(v, 16, 32);
    v += __shfl_xor(v, 8, 32);
    v += __shfl_xor(v, 4, 32);
    v += __shfl_xor(v, 2, 32);
    v += __shfl_xor(v, 1, 32);
    return v;
}

__device__ __forceinline__ float vmf_logp(float g, float kc, float c) {
    return kc * g + log1pf(-g * g) * c;
}

// Analytic stability shift: logp is concave; continuous max at
// g* = kc / (c + sqrt(c^2 + kc^2)) (cancellation-free), clamped to grid range.
// M >= max over grid, so exp(logp - M) <= 1; M cancels exactly in cdf/cdf[-1].
__device__ __forceinline__ float vmf_shift(float kc, float c, float g0, float gN) {
    float gs = kc / (c + sqrtf(c * c + kc * kc));
    gs = fminf(fmaxf(gs, g0), gN);
    return vmf_logp(gs, kc, c);
}

// ---------------- K2a: per-block sums of exp(logp - M) ----------------
__global__ void k_partials(const float* __restrict__ xg,
                           const float* __restrict__ kappa_p,
                           const int* __restrict__ dims_p,
                           int n, float* __restrict__ part) {
    const float kc = fmaxf(kappa_p[0], 0.1f);
    const float c  = (float)(dims_p[0] - 3) * 0.5f;
    const float M  = vmf_shift(kc, c, xg[0], xg[n - 1]);
    const int i = blockIdx.x * SCAN_BLK + threadIdx.x;
    float e = 0.0f;
    if (i < n) e = expf(vmf_logp(xg[i], kc, c) - M);
    const int lane = threadIdx.x & 31, wid = threadIdx.x >> 5;
    // wave inclusive scan (identical op sequence to K2c so totals match bitwise)
    float s = e;
    for (int o = 1; o < 32; o <<= 1) {
        float t = __shfl_up(s, (unsigned)o, 32);
        if (lane >= o) s += t;
    }
    __shared__ float wsum[32];
    if (lane == 31) wsum[wid] = s;
    __syncthreads();
    if (threadIdx.x == 0) {
        float acc = 0.0f;
        for (int w = 0; w < 32; ++w) acc += wsum[w];   // serial order
        part[blockIdx.x] = acc;
    }
}

// ---------------- K2b: exclusive scan of block sums (LDS-staged, serial assoc.) ----
__global__ void k_scan_parts(float* __restrict__ part, int nparts,
                             float* __restrict__ scal,
                             const float* __restrict__ kappa_p,
                             float* __restrict__ out_tail) {
    __shared__ float sp[MAX_PARTS];
    const int t = threadIdx.x;
    for (int i = t; i < nparts; i += blockDim.x) sp[i] = part[i];  // parallel load
    __syncthreads();
    if (t == 0) {
        float run = 0.0f;
        for (int i = 0; i < nparts; ++i) {       // serial order (matches K2a/K2c)
            float p = sp[i];
            sp[i] = run;
            run += p;
        }
        scal[0] = run;                           // == unnormalized cdf[n-1]
        out_tail[0] = fmaxf(kappa_p[0], 0.1f);   // second tuple output: kappa_c
    }
    __syncthreads();
    for (int i = t; i < nparts; i += blockDim.x) part[i] = sp[i];  // parallel store
}

// ---------------- K2c: write normalized cdf (== cumsum(e)/total) ----------------
__global__ void k_scan_write(const float* __restrict__ xg,
                             const float* __restrict__ kappa_p,
                             const int* __restrict__ dims_p,
                             int n, const float* __restrict__ scal,
                             const float* __restrict__ part,
                             float* __restrict__ cdf) {
    const float kc = fmaxf(kappa_p[0], 0.1f);
    const float c  = (float)(dims_p[0] - 3) * 0.5f;
    const float M  = vmf_shift(kc, c, xg[0], xg[n - 1]);
    const float tot = scal[0];
    const int i = blockIdx.x * SCAN_BLK + threadIdx.x;
    float e = 0.0f;
    if (i < n) e = expf(vmf_logp(xg[i], kc, c) - M);
    const int lane = threadIdx.x & 31, wid = threadIdx.x >> 5;
    float s = e;
    for (int o = 1; o < 32; o <<= 1) {
        float t = __shfl_up(s, (unsigned)o, 32);
        if (lane >= o) s += t;
    }
    __shared__ float wsum[32];
    __shared__ float woff[32];
    if (lane == 31) wsum[wid] = s;
    __syncthreads();
    if (threadIdx.x == 0) {
        float acc = 0.0f;
        for (int w = 0; w < 32; ++w) { woff[w] = acc; acc += wsum[w]; } // serial
    }
    __syncthreads();
    // last element: (part[last] + full-block serial total) / tot == 1.0f exactly
    if (i < n) cdf[i] = (part[blockIdx.x] + woff[wid] + s) / tot;
}

// ---------------- K3: bucket acceleration table ----------------
__global__ void k_tab(const float* __restrict__ cdf, int n,
                      int* __restrict__ tab) {
    const int k = blockIdx.x * blockDim.x + threadIdx.x;
    if (k > TAB_K) return;
    const float q = (float)k / (float)TAB_K;     // exact: K is a power of 2
    int lo = 0, hi = n;
    while (lo < hi) {
        int mid = (lo + hi) >> 1;
        if (cdf[mid] >= q) hi = mid; else lo = mid + 1;
    }
    tab[k] = min(lo, n - 1);
}

// ---------------- per-element inverse-CDF (np.interp semantics) ----------------
__device__ __forceinline__ float vmf_interp(float q,
                                            const float* __restrict__ cdf,
                                            const float* __restrict__ xg,
                                            const int* __restrict__ tab,
                                            float cdf0, float cdfN,
                                            float xg0, float xgN) {
    const float fK   = (float)TAB_K;
    const float invK = 1.0f / fK;
    int b = (int)(q * fK);
    b = max(0, min(b, TAB_K - 1));
    // repair product-rounding so that  b/K <= q < (b+1)/K
    if ((float)b * invK > q)               b--;
    else if ((float)(b + 1) * invK <= q)   b++;
    b = max(0, min(b, TAB_K - 1));
    int lo = tab[b], hi = tab[b + 1];
    while (lo < hi) {
        int mid = (lo + hi) >> 1;
        if (cdf[mid] >= q) hi = mid; else lo = mid + 1;
    }
    const int ridx = lo;
    const int lidx = max(ridx - 1, 0);
    const float xl = cdf[lidx], xr = cdf[ridx];
    const float yl = xg[lidx],  yr = xg[ridx];
    float den = xr - xl;
    den = (den == 0.0f) ? 1.0f : den;
    float v = yl + (q - xl) * (yr - yl) / den;
    if (q < cdf0) v = xg0;
    if (q > cdfN) v = xgN;
    return v;
}

// ---------------- K4: main fused kernel (one 512-row per block) ----------------
__global__ __launch_bounds__(128)
void k_main(const float* __restrict__ x_in,
            const float* __restrict__ nu_in,    // noise_u
            const float* __restrict__ nn_in,    // noise_n
            const float* __restrict__ xg,
            const float* __restrict__ cdf,
            const int* __restrict__ tab,
            int n, float* __restrict__ out) {
    const int t = threadIdx.x;                       // 0..127, 4 elems each
    const size_t base = (size_t)blockIdx.x * ROW_D + (size_t)t * 4;

    // warm L2 for the uniform-noise row while we stage/reduce (gfx1250 prefetch)
    __builtin_prefetch(nu_in + base, 0, 0);

#if VMF_ASYNC_LDS
    __shared__ __align__(16) float sx[ROW_D];
    __shared__ __align__(16) float sn[ROW_D];
    __builtin_amdgcn_global_load_async_to_lds_b128(
        VMF_AS1(x_in + base), VMF_AS3(&sx[t * 4]), 0, 0);
    __builtin_amdgcn_global_load_async_to_lds_b128(
        VMF_AS1(nn_in + base), VMF_AS3(&sn[t * 4]), 0, 0);
#if __has_builtin(__builtin_amdgcn_s_wait_asynccnt)
    __builtin_amdgcn_s_wait_asynccnt(0);
#else
    asm volatile("s_wait_asynccnt 0" ::: "memory");
#endif
    __syncthreads();
    const float4 x4 = *(const float4*)(sx + t * 4);
    const float4 n4 = *(const float4*)(sn + t * 4);
#else
    const float4 x4 = *(const float4*)(x_in + base);
    const float4 n4 = *(const float4*)(nn_in + base);
#endif

    __shared__ float red1[4];
    __shared__ float red2[4];
    const int lane = t & 31, wid = t >> 5;

    // dot(noise_n, x) over the row
    float d = x4.x * n4.x + x4.y * n4.y + x4.z * n4.z + x4.w * n4.w;
    d = vmf_wave_sum(d);
    if (lane == 0) red1[wid] = d;
    __syncthreads();
    const float dot = red1[0] + red1[1] + red1[2] + red1[3];

    // tangential component
    float4 nu4;
    nu4.x = n4.x - dot * x4.x;
    nu4.y = n4.y - dot * x4.y;
    nu4.z = n4.z - dot * x4.z;
    nu4.w = n4.w - dot * x4.w;

    float ns = nu4.x * nu4.x + nu4.y * nu4.y + nu4.z * nu4.z + nu4.w * nu4.w;
    ns = vmf_wave_sum(ns);
    if (lane == 0) red2[wid] = ns;
    __syncthreads();
    const float nrm = sqrtf(red2[0] + red2[1] + red2[2] + red2[3]);
    const float inv = 1.0f / fmaxf(nrm, 1e-12f);

    // inverse-CDF for 4 uniforms
    const float4 u4 = *(const float4*)(nu_in + base);
    const float cdf0 = cdf[0], cdfN = cdf[n - 1];
    const float xg0 = xg[0],  xgN = xg[n - 1];
    const float pw0 = vmf_interp(u4.x, cdf, xg, tab, cdf0, cdfN, xg0, xgN);
    const float pw1 = vmf_interp(u4.y, cdf, xg, tab, cdf0, cdfN, xg0, xgN);
    const float pw2 = vmf_interp(u4.z, cdf, xg, tab, cdf0, cdfN, xg0, xgN);
    const float pw3 = vmf_interp(u4.w, cdf, xg, tab, cdf0, cdfN, xg0, xgN);

    float4 o4;
    o4.x = pw0 * x4.x + sqrtf(1.0f - pw0 * pw0) * (nu4.x * inv);
    o4.y = pw1 * x4.y + sqrtf(1.0f - pw1 * pw1) * (nu4.y * inv);
    o4.z = pw2 * x4.z + sqrtf(1.0f - pw2 * pw2) * (nu4.z * inv);
    o4.w = pw3 * x4.w + sqrtf(1.0f - pw3 * pw3) * (nu4.w * inv);
    *(float4*)(out + base) = o4;
}

// ---------------- host launcher ----------------
extern "C" void kernel_launch(void* const* d_in, const int* in_sizes, int n_in,
                              void* d_out, int out_size, void* d_ws, size_t ws_size,
                              hipStream_t stream) {
    const float* x      = (const float*)d_in[0];
    const float* nu     = (const float*)d_in[1];   // noise_u
    const float* nn     = (const float*)d_in[2];   // noise_n
    const float* kappa  = (const float*)d_in[3];
    const float* xg     = (const float*)d_in[4];
    const int*   dims   = (const int*)d_in[5];
    float*       outF   = (float*)d_out;

    const int n      = in_sizes[4];                       // 199,999 grid points
    const int nparts = (n + SCAN_BLK - 1) / SCAN_BLK;     // 196
    const size_t rows = (size_t)in_sizes[0] / ROW_D;      // B*S

    // workspace layout
    char* ws = (char*)d_ws;
    float* cdf = (float*)ws;
    size_t o1 = (((size_t)n * 4) + 255) & ~(size_t)255;
    int* tab = (int*)(ws + o1);
    size_t o2 = o1 + ((((size_t)(TAB_K + 1) * 4) + 255) & ~(size_t)255);
    float* scal = (float*)(ws + o2);                      // [0]=total
    float* part = (float*)(ws + o2 + 256);                // nparts floats
    (void)ws_size; (void)n_in;

    k_partials<<<nparts, SCAN_BLK, 0, stream>>>(xg, kappa, dims, n, part);
    k_scan_parts<<<1, 256, 0, stream>>>(part, nparts, scal, kappa,
                                        outF + (out_size - 1));
    k_scan_write<<<nparts, SCAN_BLK, 0, stream>>>(xg, kappa, dims, n, scal, part, cdf);
    k_tab<<<(TAB_K + 256) / 256, 256, 0, stream>>>(cdf, n, tab);
    k_main<<<(unsigned)rows, 128, 0, stream>>>(x, nu, nn, xg, cdf, tab, n, outF);
}